// LocalHypergraph_29506425324178
// MI455X (gfx1250) — compile-verified
//
#include <hip/hip_runtime.h>
#include <hip/hip_bf16.h>
#include <math.h>

// Sizes: N=64, T=32, D=128, E1=128 -> NT=2048, all matrices 128-wide.
#define NTOT 2048
#define E1   128

typedef __attribute__((ext_vector_type(2))) float v2f;
typedef __attribute__((ext_vector_type(8))) float v8f;

// ---------------------------------------------------------------------------
// fp32 WMMA: D[16x16] += A[16x4] * B[4x16]   (wave32, one tile per wave)
// A layout (ISA 7.12.2): lanes 0-15 hold K={0,1}, lanes 16-31 hold K={2,3}
// B layout (mirrors A along K): VGPR0 = row k (half0) / k+2 (half1), VGPR1 = +1
// C/D: VGPR r -> row (half*8 + r), col = lane&15
// ---------------------------------------------------------------------------
__device__ __forceinline__ v8f wmma_f32(v2f a, v2f b, v8f c) {
  return __builtin_amdgcn_wmma_f32_16x16x4_f32(false, a, false, b, (short)0, c,
                                               false, false);
}

// ------------------------------ reductions ---------------------------------
template <int BS>
__device__ __forceinline__ float brsum(float v, float* sbuf) {
  int tid = threadIdx.x;
  sbuf[tid] = v;
  __syncthreads();
  for (int s = BS / 2; s > 0; s >>= 1) {
    if (tid < s) sbuf[tid] += sbuf[tid + s];
    __syncthreads();
  }
  float r = sbuf[0];
  __syncthreads();
  return r;
}

template <int BS>
__device__ __forceinline__ float brmax(float v, float* sbuf) {
  int tid = threadIdx.x;
  sbuf[tid] = v;
  __syncthreads();
  for (int s = BS / 2; s > 0; s >>= 1) {
    if (tid < s) sbuf[tid] = fmaxf(sbuf[tid], sbuf[tid + s]);
    __syncthreads();
  }
  float r = sbuf[0];
  __syncthreads();
  return r;
}

// ---------------------------------------------------------------------------
// NN GEMM: C[M,128] = A[M,K] @ B[K,128] (+epilogue).
// Block = 256 thr = 8 waves. Block tile = 32 rows x 128 cols:
//   wave w owns n-tile w (16 cols) for BOTH m-tiles (m0, m0+16), sharing one
//   B fragment across two WMMAs (2x compute per B load, 2 indep WMMAs/iter).
// Software pipelined: next k-fragment loads are issued before current WMMAs.
// EPI: 0 = store, 1 = +bias, 2 = relu(+bias), 3 = X + elu(acc) (residual)
// ---------------------------------------------------------------------------
struct NNArg {
  const float* A;
  const float* B;
  const float* bias;
  const float* X;  // residual input for EPI==3 (ld = 128)
  float* C;
  int lda, ldb, ldc, K;
};
struct NNArgs4 { NNArg g[4]; };

template <int EPI>
__global__ void k_gemm_nn(NNArgs4 args) {
  NNArg g = args.g[blockIdx.y];
  const int wave = threadIdx.x >> 5;
  const int lane = threadIdx.x & 31;
  const int half = lane >> 4;
  const int lid  = lane & 15;
  const int m0 = blockIdx.x * 32;  // two m-tiles: m0 and m0+16
  const int n0 = wave * 16;

  const size_t lda = g.lda, ldb = g.ldb;
  const float* a0p = g.A + (size_t)(m0 + lid) * lda + half * 2;
  const float* a1p = a0p + 16 * lda;
  const float* bp  = g.B + (size_t)(half * 2) * ldb + n0 + lid;  // row k+2h

  v8f acc0 = {}, acc1 = {};
  // prologue fragment (k = 0)
  v2f a0 = *(const v2f*)a0p;
  v2f a1 = *(const v2f*)a1p;
  v2f b;
  b.x = bp[0];
  b.y = bp[ldb];
  const float* bk = bp + 4 * ldb;
  const int K = g.K;
#pragma unroll 4
  for (int k = 4; k < K; k += 4) {
    // issue next fragment loads, then compute with current -> loads overlap WMMA
    v2f na0 = *(const v2f*)(a0p + k);
    v2f na1 = *(const v2f*)(a1p + k);
    v2f nb;
    nb.x = bk[0];
    nb.y = bk[ldb];
    bk += 4 * ldb;
    acc0 = wmma_f32(a0, b, acc0);
    acc1 = wmma_f32(a1, b, acc1);
    a0 = na0;
    a1 = na1;
    b = nb;
  }
  acc0 = wmma_f32(a0, b, acc0);
  acc1 = wmma_f32(a1, b, acc1);

  const size_t ldc = g.ldc;
#pragma unroll
  for (int t = 0; t < 2; ++t) {
    v8f acc = (t == 0) ? acc0 : acc1;
#pragma unroll
    for (int r = 0; r < 8; ++r) {
      const int m = m0 + t * 16 + half * 8 + r;
      const int n = n0 + lid;
      float v = acc[r];
      if (EPI == 1 || EPI == 2) v += g.bias[n];
      if (EPI == 2) v = fmaxf(v, 0.0f);
      if (EPI == 3) {
        const float e = v > 0.0f ? v : (expf(v) - 1.0f);  // elu, alpha=1
        v = g.X[(size_t)m * 128 + n] + e;
      }
      g.C[(size_t)m * ldc + n] = v;
    }
  }
}

// ---------------------------------------------------------------------------
// TN GEMM: C[128,128] = A^T @ B with A,B = [K,128] row-major. Split-K across
// the 8 waves of the block (one 16x16 tile / block), LDS cross-wave reduce.
// Pointer-increment addressing (+4 rows/step) + register double buffering.
// EPI: 0 = store, 1 = relu(+bias[n]), 2 = scale row: C[m,n] *= scale[m]
// ---------------------------------------------------------------------------
struct TNArg {
  const float* A;
  const float* B;
  const float* bias;
  const float* scale;
  float* C;
};
struct TNArgs4 { TNArg g[4]; };

template <int EPI>
__global__ void k_gemm_tn(TNArgs4 args, int K) {
  TNArg g = args.g[blockIdx.y];
  const int tile = blockIdx.x;  // 64 tiles over the 128x128 output
  const int m0 = (tile >> 3) * 16;
  const int n0 = (tile & 7) * 16;
  const int wave = threadIdx.x >> 5;
  const int lane = threadIdx.x & 31;
  const int half = lane >> 4;
  const int lid  = lane & 15;

  const int kslice = K >> 3;  // per-wave K chunk (K=2048 -> 256)
  const int kbeg = wave * kslice;
  // lane-private columns; advance 4 rows (512 floats) per k-step
  const float* Ap = g.A + (size_t)(kbeg + half * 2) * 128 + m0 + lid;
  const float* Bp = g.B + (size_t)(kbeg + half * 2) * 128 + n0 + lid;

  v8f acc = {};
  v2f a, b;
  a.x = Ap[0];
  a.y = Ap[128];
  b.x = Bp[0];
  b.y = Bp[128];
  const int steps = kslice >> 2;
#pragma unroll 4
  for (int s = 1; s < steps; ++s) {
    Ap += 512;
    Bp += 512;
    v2f na, nb;
    na.x = Ap[0];
    na.y = Ap[128];
    nb.x = Bp[0];
    nb.y = Bp[128];
    acc = wmma_f32(a, b, acc);
    a = na;
    b = nb;
  }
  acc = wmma_f32(a, b, acc);

  __shared__ float red[8][256];
#pragma unroll
  for (int r = 0; r < 8; ++r)
    red[wave][(half * 8 + r) * 16 + lid] = acc[r];
  __syncthreads();

  float s = 0.0f;
#pragma unroll
  for (int w2 = 0; w2 < 8; ++w2) s += red[w2][threadIdx.x];

  const int m = m0 + (threadIdx.x >> 4);
  const int n = n0 + (threadIdx.x & 15);
  if (EPI == 1) s = fmaxf(s + g.bias[n], 0.0f);
  if (EPI == 2) s *= g.scale[m];
  g.C[(size_t)m * 128 + n] = s;
}

// ---------------------------------------------------------------------------
// Column standardize (ddof=1) + softmax over the 2048 rows, per column.
// Emits probs H (row-major, for GEMMs) and transposed H / logH (for JSD).
// One block (256 thr) per column; monotone affine => max(v) gives max(s).
// ---------------------------------------------------------------------------
__global__ void k_colsoftmax(const float* __restrict__ Hin,
                             float* __restrict__ H, float* __restrict__ HT,
                             float* __restrict__ LHT) {
  __shared__ float sbuf[256];
  const int e = blockIdx.x, tid = threadIdx.x;
  float v[8], sum = 0.0f, sq = 0.0f, mx = -3.4e38f;
#pragma unroll
  for (int i = 0; i < 8; ++i) {
    const float x = Hin[(size_t)(i * 256 + tid) * 128 + e];
    v[i] = x;
    sum += x;
    sq += x * x;
    mx = fmaxf(mx, x);
  }
  sum = brsum<256>(sum, sbuf);
  sq  = brsum<256>(sq, sbuf);
  mx  = brmax<256>(mx, sbuf);

  const float mean = sum * (1.0f / 2048.0f);
  const float var  = (sq - 2048.0f * mean * mean) * (1.0f / 2047.0f);
  const float inv  = 1.0f / (sqrtf(var) + 1e-6f);
  const float smax = (mx - mean) * inv;

  float sv[8], es = 0.0f;
#pragma unroll
  for (int i = 0; i < 8; ++i) {
    sv[i] = (v[i] - mean) * inv - smax;
    es += expf(sv[i]);
  }
  es = brsum<256>(es, sbuf);
  const float lse = logf(es), rs = 1.0f / es;
#pragma unroll
  for (int i = 0; i < 8; ++i) {
    const int r = i * 256 + tid;
    const float p = expf(sv[i]) * rs;
    H[(size_t)r * 128 + e] = p;     // row-major (GEMM operand)
    HT[(size_t)e * 2048 + r] = p;   // coalesced for JSD
    LHT[(size_t)e * 2048 + r] = sv[i] - lse;
  }
}

// ---------------------------------------------------------------------------
// JSD between columns i,j of H: jsd = 0.5*sum_k[ hi*(lhi-lm) + hj*(lhj-lm) ],
// lm = log(0.5*(hi+hj)). One block per (i<=j) pair; writes both triangle halves.
// ---------------------------------------------------------------------------
__global__ void k_jsd(const float* __restrict__ HT, const float* __restrict__ LHT,
                      float* __restrict__ jsd) {
  __shared__ float sbuf[256];
  int p = blockIdx.x, i = 0;
  while (p >= 128 - i) { p -= 128 - i; ++i; }
  const int j = i + p;
  if (i == j) {
    if (threadIdx.x == 0) jsd[i * 128 + j] = 0.0f;
    return;
  }
  const float* Hi = HT + (size_t)i * 2048;
  const float* Hj = HT + (size_t)j * 2048;
  const float* Li = LHT + (size_t)i * 2048;
  const float* Lj = LHT + (size_t)j * 2048;
  float acc = 0.0f;
#pragma unroll 2
  for (int k = threadIdx.x; k < 2048; k += 256) {
    const float hi = Hi[k], hj = Hj[k];
    const float lm = logf(0.5f * (hi + hj));
    acc += hi * (Li[k] - lm) + hj * (Lj[k] - lm);
  }
  acc = brsum<256>(acc, sbuf);
  if (threadIdx.x == 0) {
    const float r = 0.5f * acc;
    jsd[i * 128 + j] = r;
    jsd[j * 128 + i] = r;
  }
}

// jm = column means of jsd; standardize (ddof=1); softmax -> hyperedge weights
__global__ void k_weights(const float* __restrict__ jsd, float* __restrict__ w) {
  __shared__ float sbuf[128];
  const int e = threadIdx.x;  // 128 threads
  float s = 0.0f;
  for (int i = 0; i < 128; ++i) s += jsd[i * 128 + e];
  const float jm = s * (1.0f / 128.0f);
  const float mean = brsum<128>(jm, sbuf) * (1.0f / 128.0f);
  const float d = jm - mean;
  const float var = brsum<128>(d * d, sbuf) * (1.0f / 127.0f);
  const float njm = d / (sqrtf(var) + 1e-6f);
  const float mx = brmax<128>(njm, sbuf);
  const float ex = expf(njm - mx);
  const float tot = brsum<128>(ex, sbuf);
  w[e] = ex / tot;
}

// ---------------------------------------------------------------------------
extern "C" void kernel_launch(void* const* d_in, const int* in_sizes, int n_in,
                              void* d_out, int out_size, void* d_ws,
                              size_t ws_size, hipStream_t stream) {
  auto F = [&](int i) { return (const float*)d_in[i]; };
  const float* xt = F(0);
  const float* xn = F(1);
  // it:2-5  inw:6-9  t2n:10-13  n2t:14-17  f:18-21  theta:22-23
  const float* fW1 = F(18); const float* fb1 = F(19);
  const float* fW2 = F(20); const float* fb2 = F(21);
  const float* thT = F(22); const float* thN = F(23);
  float* out = (float*)d_out;

  // workspace layout (floats)
  float* ws  = (float*)d_ws;
  float* R1  = ws;             // 4 x 128x128
  float* P   = R1 + 4 * 16384; // 4 x 128x128
  float* Hc  = P + 4 * 16384;  // 2048 x 512
  float* Hf1 = Hc + 2048 * 512;   // 2048 x 128
  float* H0  = Hf1 + 2048 * 128;  // 2048 x 128 (pre-softmax)
  float* H   = H0 + 2048 * 128;   // 2048 x 128 probs, row-major
  float* HT  = H + 2048 * 128;    // 128 x 2048
  float* LHT = HT + 2048 * 128;   // 128 x 2048
  float* JS  = LHT + 2048 * 128;  // 128 x 128
  float* W   = JS + 16384;        // 128
  float* Y   = W + 128;           // 2 x 2048x128
  float* Z   = Y + 2 * 2048 * 128;  // 2 x 128x128

  // 1) R1_b = relu(A_b^T @ W1_b + b1_b)  [128,128], K=2048, split-K TN
  {
    TNArgs4 t{};
    t.g[0] = {xt, F(2),  F(3),  nullptr, R1 + 0 * 16384};  // it  (x_time)
    t.g[1] = {xn, F(6),  F(7),  nullptr, R1 + 1 * 16384};  // inw (x_news)
    t.g[2] = {xn, F(10), F(11), nullptr, R1 + 2 * 16384};  // t2n (x_news^T)
    t.g[3] = {xt, F(14), F(15), nullptr, R1 + 3 * 16384};  // n2t (x_time^T)
    k_gemm_tn<1><<<dim3(64, 4), 256, 0, stream>>>(t, 2048);
  }
  // 2) P_b = R1_b @ W2_b + b2_b  [128,128] NN  (M=128 -> 4 m-blocks of 32)
  {
    NNArgs4 a{};
    a.g[0] = {R1 + 0 * 16384, F(4),  F(5),  nullptr, P + 0 * 16384, 128, 128, 128, 128};
    a.g[1] = {R1 + 1 * 16384, F(8),  F(9),  nullptr, P + 1 * 16384, 128, 128, 128, 128};
    a.g[2] = {R1 + 2 * 16384, F(12), F(13), nullptr, P + 2 * 16384, 128, 128, 128, 128};
    a.g[3] = {R1 + 3 * 16384, F(16), F(17), nullptr, P + 3 * 16384, 128, 128, 128, 128};
    k_gemm_nn<1><<<dim3(4, 4), 256, 0, stream>>>(a);
  }
  // 3) Hcat columns: H_time|H_news|H_t2n|H_n2t = X_b @ P_b  [2048,128] each
  {
    NNArgs4 a{};
    a.g[0] = {xt, P + 0 * 16384, nullptr, nullptr, Hc + 0,   128, 128, 512, 128};
    a.g[1] = {xn, P + 1 * 16384, nullptr, nullptr, Hc + 128, 128, 128, 512, 128};
    a.g[2] = {xt, P + 2 * 16384, nullptr, nullptr, Hc + 256, 128, 128, 512, 128};
    a.g[3] = {xn, P + 3 * 16384, nullptr, nullptr, Hc + 384, 128, 128, 512, 128};
    k_gemm_nn<0><<<dim3(64, 4), 256, 0, stream>>>(a);
  }
  // 4) Hf1 = relu(Hcat @ fW1 + fb1)  K=512
  {
    NNArgs4 a{};
    a.g[0] = {Hc, fW1, fb1, nullptr, Hf1, 512, 128, 128, 512};
    k_gemm_nn<2><<<dim3(64, 1), 256, 0, stream>>>(a);
  }
  // 5) H0 = Hf1 @ fW2 + fb2
  {
    NNArgs4 a{};
    a.g[0] = {Hf1, fW2, fb2, nullptr, H0, 128, 128, 128, 128};
    k_gemm_nn<1><<<dim3(64, 1), 256, 0, stream>>>(a);
  }
  // 6) column standardize + softmax
  k_colsoftmax<<<128, 256, 0, stream>>>(H0, H, HT, LHT);
  // 7) pairwise JSD (i<=j pairs: 128*129/2 blocks)
  k_jsd<<<8256, 256, 0, stream>>>(HT, LHT, JS);
  // 8) hyperedge weights w
  k_weights<<<1, 128, 0, stream>>>(JS, W);
  // 9) y_b = x_b @ theta_b
  {
    NNArgs4 a{};
    a.g[0] = {xt, thT, nullptr, nullptr, Y,               128, 128, 128, 128};
    a.g[1] = {xn, thN, nullptr, nullptr, Y + 2048 * 128,  128, 128, 128, 128};
    k_gemm_nn<0><<<dim3(64, 2), 256, 0, stream>>>(a);
  }
  // 10) z_b[e,d] = w[e] * sum_k H[k,e]*y_b[k,d]   (TN, K=2048, row-scale epi)
  {
    TNArgs4 t{};
    t.g[0] = {H, Y,              nullptr, W, Z};
    t.g[1] = {H, Y + 2048 * 128, nullptr, W, Z + 16384};
    k_gemm_tn<2><<<dim3(64, 2), 256, 0, stream>>>(t, 2048);
  }
  // 11) out_b = x_b + elu(H @ z_b)
  {
    NNArgs4 a{};
    a.g[0] = {H, Z,         nullptr, xt, out,              128, 128, 128, 128};
    a.g[1] = {H, Z + 16384, nullptr, xn, out + 2048 * 128, 128, 128, 128, 128};
    k_gemm_nn<3><<<dim3(64, 2), 256, 0, stream>>>(a);
  }
}